// CFGNN_9938554323124
// MI455X (gfx1250) — compile-verified
//
#include <hip/hip_runtime.h>
#include <hip/hip_bf16.h>

#define N_NODES 100000
#define N_EDGES 1600000
#define DIM     128
#define SEL2    32768          // senders (16384) + receivers (16384)
#define MLP     512
#define L2EPS   1e-12f

typedef float v2f __attribute__((ext_vector_type(2)));
typedef float v8f __attribute__((ext_vector_type(8)));

// ---------------------------------------------------------------- utilities
__global__ void cfgnn_zero_f32(float* __restrict__ p, int n) {
    int i = blockIdx.x * blockDim.x + threadIdx.x;
    if (i < n) p[i] = 0.0f;
}

// deg[dst] += 1  (f32 exact for counts < 2^24; max possible is 1.6M)
__global__ void cfgnn_deg(const long long* __restrict__ dst,
                          float* __restrict__ deg, int e) {
    int i = blockIdx.x * blockDim.x + threadIdx.x;
    if (i < e) atomicAdd(&deg[(int)dst[i]], 1.0f);
}

// deg -> rsqrt(max(deg,1)) in place
__global__ void cfgnn_dinv(float* __restrict__ deg, int n) {
    int i = blockIdx.x * blockDim.x + threadIdx.x;
    if (i < n) deg[i] = rsqrtf(fmaxf(deg[i], 1.0f));
}

// enorm[e] = dinv[src]*dinv[dst], hoisted out of the 128-wide propagate loop
__global__ void cfgnn_enorm(const long long* __restrict__ src,
                            const long long* __restrict__ dst,
                            const float* __restrict__ dinv,
                            float* __restrict__ enorm, int e) {
    int i = blockIdx.x * blockDim.x + threadIdx.x;
    if (i < e) enorm[i] = dinv[(int)src[i]] * dinv[(int)dst[i]];
}

// ------------------------------------------------------- propagation layer
// One wave per edge; lane l moves floats [4l, 4l+4) of the 128-d row.
// Gather is a coalesced 512B burst from an L2-resident table; scatter is
// 4 global_atomic_add_f32 per lane, resolved at L2.
__global__ void cfgnn_prop(const float* __restrict__ xin,
                           float* __restrict__ xout,
                           const long long* __restrict__ src,
                           const long long* __restrict__ dst,
                           const float* __restrict__ enorm, int e) {
    const int wave = blockIdx.x * (blockDim.x >> 5) + (threadIdx.x >> 5);
    const int lane = threadIdx.x & 31;
    if (wave >= e) return;
    const int s = (int)src[wave];
    const int d = (int)dst[wave];
    const float w = enorm[wave];
    const float4 v = *(const float4*)(xin + (size_t)s * DIM + lane * 4);
    float* p = xout + (size_t)d * DIM + lane * 4;
    atomicAdd(p + 0, v.x * w);
    atomicAdd(p + 1, v.y * w);
    atomicAdd(p + 2, v.z * w);
    atomicAdd(p + 3, v.w * w);
}

// --------------------------------------- gather + L2-normalize selected rows
// One wave per output row (row < 16384 -> senders, else receivers).
// zsel[row, layer*128 : layer*128+128] = x[node] / max(||x[node]||, eps)
__global__ void cfgnn_gather_norm(const float* __restrict__ xin,
                                  float* __restrict__ zsel,
                                  const long long* __restrict__ snd,
                                  const long long* __restrict__ rcv,
                                  int layer) {
    const int wave = blockIdx.x * (blockDim.x >> 5) + (threadIdx.x >> 5);
    const int lane = threadIdx.x & 31;
    if (wave >= SEL2) return;
    const long long node = (wave < 16384) ? snd[wave] : rcv[wave - 16384];
    const float4 v = *(const float4*)(xin + (size_t)node * DIM + lane * 4);
    float ss = v.x * v.x + v.y * v.y + v.z * v.z + v.w * v.w;
    #pragma unroll
    for (int m = 16; m >= 1; m >>= 1) ss += __shfl_xor(ss, m, 32);
    const float inv = 1.0f / fmaxf(sqrtf(ss), L2EPS);
    float4 o;
    o.x = v.x * inv; o.y = v.y * inv; o.z = v.z * inv; o.w = v.w * inv;
    *(float4*)(zsel + (size_t)wave * MLP + layer * DIM + lane * 4) = o;
}

// ----------------------------------------------------------- WMMA f32 GEMM
// out[32768,512] = zsel[32768,512] @ W^T + b, one 16x16 C tile per wave.
// V_WMMA_F32_16X16X4_F32: A 16x4 (2 VGPRs: lanes 0-15 K={k,k+1},
// lanes 16-31 K={k+2,k+3}), B 4x16 mirrored, C 16x16 f32 (8 VGPRs,
// M = 8*half + vgpr, N = lane&15). B[k][n] = W[n][k] -> contiguous W-row loads.
__global__ void cfgnn_gemm_wmma(const float* __restrict__ Z,
                                const float* __restrict__ W,
                                const float* __restrict__ bias,
                                float* __restrict__ out) {
    const int wave  = blockIdx.x * (blockDim.x >> 5) + (threadIdx.x >> 5);
    const int lane  = threadIdx.x & 31;
    const int mTile = wave >> 5;          // 2048 M-tiles
    const int nTile = wave & 31;          // 32 N-tiles
    const int half  = lane >> 4;
    const int l16   = lane & 15;
    const float* Aptr = Z + (size_t)(mTile * 16 + l16) * MLP + 2 * half;
    const float* Bptr = W + (size_t)(nTile * 16 + l16) * MLP + 2 * half;
    v8f c = {};
    #pragma unroll 8
    for (int k = 0; k < MLP; k += 4) {
        v2f a = *(const v2f*)(Aptr + k);
        v2f b = *(const v2f*)(Bptr + k);
        c = __builtin_amdgcn_wmma_f32_16x16x4_f32(
                false, a, false, b, (short)0, c, false, false);
    }
    const int col = nTile * 16 + l16;
    const float bb = bias[col];
    #pragma unroll
    for (int i = 0; i < 8; ++i) {
        const int row = mTile * 16 + half * 8 + i;
        out[(size_t)row * MLP + col] = c[i] + bb;
    }
}

// ---------------------------------------------------------------- launcher
extern "C" void kernel_launch(void* const* d_in, const int* in_sizes, int n_in,
                              void* d_out, int out_size, void* d_ws, size_t ws_size,
                              hipStream_t stream) {
    const float*     emb  = (const float*)d_in[0];
    const long long* edge = (const long long*)d_in[1];   // int64 [2, E]
    const long long* snd  = (const long long*)d_in[2];   // int64 [16384]
    const long long* rcv  = (const long long*)d_in[3];   // int64 [16384]
    const float*     W    = (const float*)d_in[4];       // [512, 512]
    const float*     bias = (const float*)d_in[5];       // [512]
    float*           out  = (float*)d_out;               // [32768, 512]

    const long long* src = edge;
    const long long* dst = edge + N_EDGES;

    // workspace layout (floats): dinv | enorm | xa | xb | zsel  (~176 MB)
    float* ws    = (float*)d_ws;
    float* dinv  = ws;                      // 100000 (padded to 102400)
    float* enorm = dinv + 102400;           // 1,600,000
    float* xa    = enorm + N_EDGES;         // 12,800,000
    float* xb    = xa + (size_t)N_NODES * DIM;
    float* zsel  = xb + (size_t)N_NODES * DIM;   // 16,777,216

    const int T = 256;
    const int WPB = T / 32;                               // waves per block
    const int nodeBlk = (N_NODES + T - 1) / T;
    const int edgeBlk = (N_EDGES + T - 1) / T;
    const int xBlk    = (N_NODES * DIM + T - 1) / T;
    const int propBlk = (N_EDGES + WPB - 1) / WPB;        // 1 wave / edge
    const int gathBlk = SEL2 / WPB;                       // 1 wave / sel row
    const int gemmBlk = (SEL2 / 16) * (MLP / 16) / WPB;   // 1 wave / C tile

    // degree -> dinv -> per-edge norm
    cfgnn_zero_f32<<<nodeBlk, T, 0, stream>>>(dinv, N_NODES);
    cfgnn_deg<<<edgeBlk, T, 0, stream>>>(dst, dinv, N_EDGES);
    cfgnn_dinv<<<nodeBlk, T, 0, stream>>>(dinv, N_NODES);
    cfgnn_enorm<<<edgeBlk, T, 0, stream>>>(src, dst, dinv, enorm, N_EDGES);

    // layer 0: normalized gather straight from emb
    cfgnn_gather_norm<<<gathBlk, T, 0, stream>>>(emb, zsel, snd, rcv, 0);

    // layer 1: emb -> xa
    cfgnn_zero_f32<<<xBlk, T, 0, stream>>>(xa, N_NODES * DIM);
    cfgnn_prop<<<propBlk, T, 0, stream>>>(emb, xa, src, dst, enorm, N_EDGES);
    cfgnn_gather_norm<<<gathBlk, T, 0, stream>>>(xa, zsel, snd, rcv, 1);

    // layer 2: xa -> xb
    cfgnn_zero_f32<<<xBlk, T, 0, stream>>>(xb, N_NODES * DIM);
    cfgnn_prop<<<propBlk, T, 0, stream>>>(xa, xb, src, dst, enorm, N_EDGES);
    cfgnn_gather_norm<<<gathBlk, T, 0, stream>>>(xb, zsel, snd, rcv, 2);

    // layer 3: xb -> xa (layer-1 data in xa already consumed)
    cfgnn_zero_f32<<<xBlk, T, 0, stream>>>(xa, N_NODES * DIM);
    cfgnn_prop<<<propBlk, T, 0, stream>>>(xb, xa, src, dst, enorm, N_EDGES);
    cfgnn_gather_norm<<<gathBlk, T, 0, stream>>>(xa, zsel, snd, rcv, 3);

    // post MLP on selected rows only: 32768 x 512 x 512 f32 WMMA GEMM
    cfgnn_gemm_wmma<<<gemmBlk, T, 0, stream>>>(zsel, W, bias, out);
}